// RaggedMatMul_26250840113178
// MI455X (gfx1250) — compile-verified
//
#include <hip/hip_runtime.h>

// CDNA5 / gfx1250 ragged batched GEMM:
//   out[t,:] = values[t,:] (1x64) @ mats[seg[t]] (64x64),  seg sorted.
// Memory-bound (~406 MB @ 23.3 TB/s ~ 17.5us); fp32 WMMA 16x16x4 supplies
// more than the ~370 TFLOP/s needed, so keep everything fp32.

typedef __attribute__((ext_vector_type(2))) float v2f;
typedef __attribute__((ext_vector_type(8))) float v8f;

#define DIM 64
#define WAVES_PER_BLOCK 8

__global__ __launch_bounds__(256) void ragged_mm_wmma(
    const float* __restrict__ values,  // [T, 64]
    const float* __restrict__ mats,    // [B, 64, 64]
    const int*   __restrict__ seg,     // [T], sorted
    float*       __restrict__ out,     // [T, 64]
    int T)
{
    const int lane  = threadIdx.x & 31;
    const int waveI = threadIdx.x >> 5;
    const int tile  = blockIdx.x * WAVES_PER_BLOCK + waveI;
    const int rowBase = tile * 16;
    if (rowBase >= T) return;                 // wave-uniform

    const int s0 = seg[rowBase];
    const int s1 = seg[rowBase + 15];

    if (s0 == s1) {                           // wave-uniform: EXEC stays all-1s
        // ---- WMMA path: 16 rows x 64 cols, K-loop of 16 steps of 4 ----
        const float* Arows = values + (size_t)rowBase * DIM;
        const float* Bm    = mats   + (size_t)s0 * DIM * DIM;

        // A 16x4 f32 layout: lane L -> row M = L%16; VGPR0/1 -> K = 2*(L/16)+{0,1}
        const int mRow = lane & 15;
        const int kHi  = (lane >> 4) << 1;    // 0 or 2
        // B 4x16 f32 layout (mirrors C/D): lane L -> col N = L%16;
        //   VGPR v -> K-row = 2*(L/16) + v
        const int nCol = lane & 15;

        v8f acc0 = {}, acc1 = {}, acc2 = {}, acc3 = {};

        for (int k0 = 0; k0 < DIM; k0 += 4) {
            // A fragment: two consecutive K values -> single 8-byte load
            v2f a = *(const v2f*)(Arows + (size_t)mRow * DIM + (k0 + kHi));

            const float* Brow0 = Bm + (size_t)(k0 + kHi) * DIM;  // K-row for VGPR0
            const float* Brow1 = Brow0 + DIM;                    // K-row for VGPR1

            v2f b0, b1, b2, b3;                // four 16-col tiles of B
            b0.x = Brow0[nCol];       b0.y = Brow1[nCol];
            b1.x = Brow0[16 + nCol];  b1.y = Brow1[16 + nCol];
            b2.x = Brow0[32 + nCol];  b2.y = Brow1[32 + nCol];
            b3.x = Brow0[48 + nCol];  b3.y = Brow1[48 + nCol];

            // 8 args: (neg_a, A, neg_b, B, c_mod, C, reuse_a, reuse_b)
            acc0 = __builtin_amdgcn_wmma_f32_16x16x4_f32(false, a, false, b0, (short)0, acc0, false, false);
            acc1 = __builtin_amdgcn_wmma_f32_16x16x4_f32(false, a, false, b1, (short)0, acc1, false, false);
            acc2 = __builtin_amdgcn_wmma_f32_16x16x4_f32(false, a, false, b2, (short)0, acc2, false, false);
            acc3 = __builtin_amdgcn_wmma_f32_16x16x4_f32(false, a, false, b3, (short)0, acc3, false, false);
        }

        // C/D 16x16 f32 layout: lane L -> col N = L%16; VGPR v -> row v + 8*(L/16)
        const int rOff = (lane >> 4) * 8;
        float* Orows = out + (size_t)rowBase * DIM;
        #pragma unroll
        for (int v = 0; v < 8; ++v) {
            float* orow = Orows + (size_t)(rOff + v) * DIM + nCol;
            orow[0]  = acc0[v];
            orow[16] = acc1[v];
            orow[32] = acc2[v];
            orow[48] = acc3[v];
        }
    } else {
        // ---- Segment-boundary tile (<=15 in the entire grid): VALU fallback ----
        for (int j = 0; j < 32; ++j) {
            int idx = lane + 32 * j;          // 0..1023 covers 16x64 outputs
            int r   = idx >> 6;
            int c   = idx & 63;
            int row = rowBase + r;
            const float* av = values + (size_t)row * DIM;
            const float* bm = mats + (size_t)seg[row] * DIM * DIM;
            float acc = 0.f;
            #pragma unroll 8
            for (int d = 0; d < DIM; ++d)
                acc = fmaf(av[d], bm[(size_t)d * DIM + c], acc);
            out[(size_t)row * DIM + c] = acc;
        }
    }
}

extern "C" void kernel_launch(void* const* d_in, const int* in_sizes, int n_in,
                              void* d_out, int out_size, void* d_ws, size_t ws_size,
                              hipStream_t stream) {
    const float* values = (const float*)d_in[0];   // [T, 64] fp32
    const float* mats   = (const float*)d_in[1];   // [B, 64, 64] fp32
    const int*   seg    = (const int*)d_in[2];     // [T] int32, sorted
    float*       out    = (float*)d_out;           // [T, 64] fp32

    const int T = in_sizes[0] / DIM;
    const int numTiles = (T + 15) / 16;
    const int blocks = (numTiles + WAVES_PER_BLOCK - 1) / WAVES_PER_BLOCK;

    ragged_mm_wmma<<<blocks, 256, 0, stream>>>(values, mats, seg, out, T);
}